// GCRNII_16106127360500
// MI455X (gfx1250) — compile-verified
//
#include <hip/hip_runtime.h>
#include <cmath>

typedef __attribute__((ext_vector_type(2))) float v2f;
typedef __attribute__((ext_vector_type(8))) float v8f;

#define NND   50000
#define NED   800000
#define DIN   256
#define DH    128
#define ALPHA 0.1f
#define BNEPS 1e-5f

// ---------------------------------------------------------------------------
// Degree / normalization
// ---------------------------------------------------------------------------
__global__ void k_deg_init(float* deg) {
    int i = blockIdx.x * blockDim.x + threadIdx.x;
    if (i < NND) deg[i] = 1.0f;  // self loop
}
__global__ void k_deg_edges(const int* __restrict__ dst, float* deg) {
    int e = blockIdx.x * blockDim.x + threadIdx.x;
    if (e < NED) atomicAdd(&deg[dst[e]], 1.0f);
}
__global__ void k_deg_rsqrt(float* deg) {
    int i = blockIdx.x * blockDim.x + threadIdx.x;
    if (i < NND) deg[i] = rsqrtf(fmaxf(deg[i], 1.0f));
}

// ---------------------------------------------------------------------------
// FP32 WMMA GEMM:  out[N, 128] = in[N, K] @ W[K, 128]   (+ optional blend)
// mode 0: out = acc
// mode 1: out = (1-beta)*in + beta*acc    (in-place, out == in, K == 128)
//
// One wave computes a 16-row x 128-col slab: 8 accumulator tiles, A reused 8x.
// W is staged in LDS in K-PAIR-INTERLEAVED layout:
//   sW2[(k/2)*128 + col] = { W[k][col], W[k+1][col] }   (k even)
// so each lane's B fragment (rows k,k+1 at one col) is a single aligned
// ds_load_b64 straight into an even VGPR pair -> no repack movs.
// ---------------------------------------------------------------------------
template <int K>
__global__ __launch_bounds__(256)
void k_gemm_wmma(const float* in, const float* __restrict__ W,
                 float* out, int mode, float beta) {
    extern __shared__ float2 sW2[];  // (K/2) * DH float2
    const int tid = threadIdx.x;

    // cooperative swizzling load of W into LDS
    {
        const int totalIter = (K / 2) * (DH / 4);   // each iter covers 2 rows x 4 cols
        for (int i = tid; i < totalIter; i += 256) {
            const int kp = i / (DH / 4);        // k-pair index
            const int c4 = i % (DH / 4);        // col group of 4
            const float4 a0 = *(const float4*)(W + (size_t)(2 * kp)     * DH + c4 * 4);
            const float4 a1 = *(const float4*)(W + (size_t)(2 * kp + 1) * DH + c4 * 4);
            float4* dst4 = (float4*)(sW2 + (size_t)kp * DH + c4 * 4);
            float4 f0; f0.x = a0.x; f0.y = a1.x; f0.z = a0.y; f0.w = a1.y;
            float4 f1; f1.x = a0.z; f1.y = a1.z; f1.z = a0.w; f1.w = a1.w;
            dst4[0] = f0;
            dst4[1] = f1;
        }
    }
    __syncthreads();

    const int wave = tid >> 5;
    const int lane = tid & 31;
    const int tileRow = blockIdx.x * 8 + wave;   // per-wave uniform
    if (tileRow >= NND / 16) return;             // whole wave exits: EXEC stays all-1s
    const int m0 = tileRow * 16;
    const int r    = lane & 15;
    const bool hi  = lane >= 16;
    const int koff = hi ? 2 : 0;
    const float* arow = in + (size_t)(m0 + r) * K;

    v8f c[8] = {};
    for (int k0 = 0; k0 < K; k0 += 4) {
        // A fragment: 16x4 f32; lanes 0-15 hold K=k0..k0+1, lanes 16-31 K=k0+2..k0+3
        const float2 a2 = *(const float2*)(arow + k0 + koff);
        v2f a; a.x = a2.x; a.y = a2.y;
        const int pairRow = (k0 >> 1) + (hi ? 1 : 0);
        const float2* brow = sW2 + (size_t)pairRow * DH + r;
#pragma unroll
        for (int t = 0; t < 8; ++t) {
            const float2 b2 = brow[t * 16];     // single ds_load_b64
            v2f b; b.x = b2.x; b.y = b2.y;
            c[t] = __builtin_amdgcn_wmma_f32_16x16x4_f32(
                false, a, false, b, (short)0, c[t], false, false);
        }
    }

    // Epilogue. C/D layout: VGPR v, lanes 0-15 -> M=v, lanes 16-31 -> M=v+8.
#pragma unroll
    for (int t = 0; t < 8; ++t) {
        const int col = t * 16 + r;
#pragma unroll
        for (int v = 0; v < 8; ++v) {
            const int row = m0 + v + (hi ? 8 : 0);
            const size_t o = (size_t)row * DH + col;
            float val = c[t][v];
            if (mode != 0) val = (1.0f - beta) * in[o] + beta * val;  // K==DH here
            out[o] = val;
        }
    }
}

// ---------------------------------------------------------------------------
// GCNConv aggregation:  z = D^{-1/2}(A+I)D^{-1/2} (xW) + b
// ---------------------------------------------------------------------------
__global__ __launch_bounds__(256)
void k_gcn_self_bias(const float* __restrict__ xw, const float* __restrict__ dinv,
                     const float* __restrict__ b, float* __restrict__ z) {
    int i = blockIdx.x * blockDim.x + threadIdx.x;  // N*32 threads, 4 ch each
    int node = i >> 5, l = i & 31;
    float w = dinv[node] * dinv[node];
    float4 v  = *(const float4*)(xw + (size_t)node * DH + l * 4);
    float4 bb = *(const float4*)(b + l * 4);
    float4 o;
    o.x = w * v.x + bb.x; o.y = w * v.y + bb.y;
    o.z = w * v.z + bb.z; o.w = w * v.w + bb.w;
    *(float4*)(z + (size_t)node * DH + l * 4) = o;
}

__global__ __launch_bounds__(256)
void k_gcn_edge(const int* __restrict__ src, const int* __restrict__ dst,
                const float* __restrict__ dinv, const float* __restrict__ xw,
                float* z) {
    int e = blockIdx.x * 8 + (threadIdx.x >> 5);  // 1 wave per edge
    int l = threadIdx.x & 31;                     // 4 channels per lane
    int s = src[e], d = dst[e];
    float w = dinv[s] * dinv[d];
    const float4 v = *(const float4*)(xw + (size_t)s * DH + l * 4);
    float* p = z + (size_t)d * DH + l * 4;
    atomicAdd(p + 0, w * v.x);
    atomicAdd(p + 1, w * v.y);
    atomicAdd(p + 2, w * v.z);
    atomicAdd(p + 3, w * v.w);
}

// ---------------------------------------------------------------------------
// GCN2 aggregation:  h = alpha*x0 ; h[dst] += (1-alpha)*z[src]
// ---------------------------------------------------------------------------
__global__ __launch_bounds__(256)
void k_h_init(const float* __restrict__ x0, float* __restrict__ h) {
    size_t i = (size_t)blockIdx.x * blockDim.x + threadIdx.x;  // N*32 threads
    float4 v = *(const float4*)(x0 + i * 4);
    float4 o; o.x = ALPHA * v.x; o.y = ALPHA * v.y; o.z = ALPHA * v.z; o.w = ALPHA * v.w;
    *(float4*)(h + i * 4) = o;
}

__global__ __launch_bounds__(256)
void k_gcn2_edge(const int* __restrict__ src, const int* __restrict__ dst,
                 const float* __restrict__ z, float* h) {
    int e = blockIdx.x * 8 + (threadIdx.x >> 5);
    int l = threadIdx.x & 31;
    int s = src[e], d = dst[e];
    const float4 v = *(const float4*)(z + (size_t)s * DH + l * 4);
    float* p = h + (size_t)d * DH + l * 4;
    const float w = 1.0f - ALPHA;
    atomicAdd(p + 0, w * v.x);
    atomicAdd(p + 1, w * v.y);
    atomicAdd(p + 2, w * v.z);
    atomicAdd(p + 3, w * v.w);
}

// ---------------------------------------------------------------------------
// BatchNorm (training mode, biased var) + ReLU
// stats layout: [0..127]=sum [128..255]=sumsq [256..383]=mean [384..511]=invstd
// ---------------------------------------------------------------------------
__global__ void k_stats_zero(float* stats) { stats[threadIdx.x] = 0.0f; }

__global__ __launch_bounds__(256)
void k_bn_stats(const float* __restrict__ z, float* stats, int rowsPerBlock) {
    __shared__ float ssum[DH], ssq[DH];
    int tid = threadIdx.x;
    if (tid < DH) { ssum[tid] = 0.0f; ssq[tid] = 0.0f; }
    __syncthreads();
    int c = tid & (DH - 1);
    int rsub = tid >> 7;  // 0..1
    int base = blockIdx.x * rowsPerBlock;
    int rend = base + rowsPerBlock; if (rend > NND) rend = NND;
    float s = 0.0f, q = 0.0f;
    for (int r = base + rsub; r < rend; r += 2) {
        float v = z[(size_t)r * DH + c];
        s += v; q += v * v;
    }
    atomicAdd(&ssum[c], s);  // LDS f32 atomic (ds_add_f32)
    atomicAdd(&ssq[c], q);
    __syncthreads();
    if (tid < DH) {
        atomicAdd(&stats[tid], ssum[tid]);
        atomicAdd(&stats[DH + tid], ssq[tid]);
    }
}

__global__ void k_bn_final(float* stats) {
    int c = threadIdx.x;  // 128 threads
    float invN = 1.0f / (float)NND;
    float mean = stats[c] * invN;
    float var  = stats[DH + c] * invN - mean * mean;
    stats[2 * DH + c] = mean;
    stats[3 * DH + c] = rsqrtf(var + BNEPS);
}

__global__ __launch_bounds__(256)
void k_bn_apply(float* z, const float* __restrict__ stats,
                const float* __restrict__ gamma, const float* __restrict__ beta) {
    size_t i = (size_t)blockIdx.x * blockDim.x + threadIdx.x;  // N*DH threads
    int c = (int)(i & (DH - 1));
    float v = z[i];
    v = (v - stats[2 * DH + c]) * stats[3 * DH + c] * gamma[c] + beta[c];
    z[i] = fmaxf(v, 0.0f);
}

// ---------------------------------------------------------------------------
extern "C" void kernel_launch(void* const* d_in, const int* in_sizes, int n_in,
                              void* d_out, int out_size, void* d_ws, size_t ws_size,
                              hipStream_t stream) {
    const float* x        = (const float*)d_in[0];
    const int*   edge     = (const int*)d_in[1];     // [2, E]
    const float* W_lin    = (const float*)d_in[2];   // [256,128]
    const float* b_lin    = (const float*)d_in[3];   // [128]
    const float* W1s      = (const float*)d_in[4];   // [5,128,128]
    const float* bn_gamma = (const float*)d_in[5];   // [4,128]
    const float* bn_beta  = (const float*)d_in[6];   // [4,128]
    float* out = (float*)d_out;

    float* ws    = (float*)d_ws;
    float* dinv  = ws;                                  // N
    float* stats = ws + NND;                            // 512
    float* bufA  = stats + 512;                         // N*128
    float* bufB  = bufA + (size_t)NND * DH;             // N*128
    float* x0    = bufB + (size_t)NND * DH;             // N*128

    const int* src = edge;
    const int* dst = edge + NED;

    const int nBlkN   = (NND + 255) / 256;
    const int nBlkE   = (NED + 255) / 256;
    const int nBlkVec = NND * 32 / 256;        // 6250 (exact)
    const int nBlkEdg = NED / 8;               // 100000 (exact)
    const int nBlkGemm = (NND / 16 + 7) / 8;   // 391

    // 1) degree -> dinv
    k_deg_init<<<nBlkN, 256, 0, stream>>>(dinv);
    k_deg_edges<<<nBlkE, 256, 0, stream>>>(dst, dinv);
    k_deg_rsqrt<<<nBlkN, 256, 0, stream>>>(dinv);

    // 2) xw = x @ W_lin   -> bufA   (fp32 WMMA, W in LDS: 128 KB)
    k_gemm_wmma<DIN><<<nBlkGemm, 256, DIN * DH * sizeof(float), stream>>>(
        x, W_lin, bufA, 0, 0.0f);

    // 3) GCNConv aggregation -> bufB
    k_gcn_self_bias<<<nBlkVec, 256, 0, stream>>>(bufA, dinv, b_lin, bufB);
    k_gcn_edge<<<nBlkEdg, 256, 0, stream>>>(src, dst, dinv, bufA, bufB);

    // 4) x0 = z
    hipMemcpyAsync(x0, bufB, (size_t)NND * DH * sizeof(float),
                   hipMemcpyDeviceToDevice, stream);

    float* z = bufB;
    float* h = bufA;
    for (int i = 0; i < 5; ++i) {
        const float beta = logf(0.5f / (float)(i + 1) + 1.0f);
        k_h_init<<<nBlkVec, 256, 0, stream>>>(x0, h);
        k_gcn2_edge<<<nBlkEdg, 256, 0, stream>>>(src, dst, z, h);
        // in-place: h <- (1-beta)*h + beta*(h @ W1[i])   (W1 in LDS: 64 KB)
        k_gemm_wmma<DH><<<nBlkGemm, 256, DH * DH * sizeof(float), stream>>>(
            h, W1s + (size_t)i * DH * DH, h, 1, beta);
        if (i < 4) {
            k_stats_zero<<<1, 256, 0, stream>>>(stats);
            k_bn_stats<<<(NND + 63) / 64, 256, 0, stream>>>(h, stats, 64);
            k_bn_final<<<1, DH, 0, stream>>>(stats);
            k_bn_apply<<<NND * DH / 256, 256, 0, stream>>>(
                h, stats, bn_gamma + i * DH, bn_beta + i * DH);
        }
        float* t = z; z = h; h = t;
    }

    hipMemcpyAsync(out, z, (size_t)NND * DH * sizeof(float),
                   hipMemcpyDeviceToDevice, stream);
}